// HACK_31490700214642
// MI455X (gfx1250) — compile-verified
//
#include <hip/hip_runtime.h>
#include <hip/hip_bf16.h>

#define NB 8
#define NVTX 14062
#define N3 42186      // NVTX*3
#define KP 128        // K=127 padded
#define NBATCH 256

typedef __attribute__((ext_vector_type(2))) float v2f;
typedef __attribute__((ext_vector_type(8))) float v8f;

__device__ __forceinline__ void mat4mul(float* __restrict__ d,
                                        const float* __restrict__ a,
                                        const float* __restrict__ b) {
#pragma unroll
    for (int i = 0; i < 4; ++i)
#pragma unroll
        for (int j = 0; j < 4; ++j) {
            float s = 0.f;
#pragma unroll
            for (int k = 0; k < 4; ++k) s = fmaf(a[i * 4 + k], b[k * 4 + j], s);
            d[i * 4 + j] = s;
        }
}

// inverse of rigid [R t; 0 1] = [R^T -R^T t; 0 1]
__device__ __forceinline__ void rigid_inv(float* __restrict__ d,
                                          const float* __restrict__ m) {
    d[0] = m[0]; d[1] = m[4]; d[2]  = m[8];
    d[4] = m[1]; d[5] = m[5]; d[6]  = m[9];
    d[8] = m[2]; d[9] = m[6]; d[10] = m[10];
    d[3]  = -(m[0] * m[3] + m[4] * m[7] + m[8]  * m[11]);
    d[7]  = -(m[1] * m[3] + m[5] * m[7] + m[9]  * m[11]);
    d[11] = -(m[2] * m[3] + m[6] * m[7] + m[10] * m[11]);
    d[12] = 0.f; d[13] = 0.f; d[14] = 0.f; d[15] = 1.f;
}

// One block of 256 threads; thread b handles batch b.
__global__ void k_precompute(const float* __restrict__ theta, // [256][8][3]
                             const float* __restrict__ bsw,   // [256][55]
                             const float* __restrict__ L2P,   // [8][4][4]
                             float* __restrict__ A,           // [256][128]
                             float* __restrict__ Mws)         // [256][8][12]
{
    int b = threadIdx.x;

    float R[NB][9];
#pragma unroll
    for (int n = 0; n < NB; ++n) {
        float rx = theta[(b * NB + n) * 3 + 0];
        float ry = theta[(b * NB + n) * 3 + 1];
        float rz = theta[(b * NB + n) * 3 + 2];
        float ang = sqrtf(rx * rx + ry * ry + rz * rz + 1e-12f);
        float inv = 1.0f / ang;
        float kx = rx * inv, ky = ry * inv, kz = rz * inv;
        float s = sinf(ang), c = cosf(ang), omc = 1.0f - c;
        R[n][0] = 1.f - omc * (ky * ky + kz * kz);
        R[n][1] = -s * kz + omc * (kx * ky);
        R[n][2] =  s * ky + omc * (kx * kz);
        R[n][3] =  s * kz + omc * (kx * ky);
        R[n][4] = 1.f - omc * (kx * kx + kz * kz);
        R[n][5] = -s * kx + omc * (ky * kz);
        R[n][6] = -s * ky + omc * (kx * kz);
        R[n][7] =  s * kx + omc * (ky * kz);
        R[n][8] = 1.f - omc * (kx * kx + ky * ky);
    }

    // A row: 72 theta_zero (bone0 subtracts itself -> zeros) + 55 bsw + pad
    float* Ar = A + b * KP;
#pragma unroll
    for (int n = 0; n < NB; ++n)
#pragma unroll
        for (int e = 0; e < 9; ++e) {
            float sub = (n == 0) ? R[0][e] : ((e == 0 || e == 4 || e == 8) ? 1.f : 0.f);
            Ar[n * 9 + e] = R[n][e] - sub;
        }
    for (int j = 0; j < 55; ++j) Ar[72 + j] = bsw[b * 55 + j];
    Ar[127] = 0.f;

    // FK chain + inverse bind + W2L2pWs
    float prev_pose[16], prev_rest[16];
#pragma unroll
    for (int n = 0; n < NB; ++n) {
        float loc[16] = {R[n][0], R[n][1], R[n][2], 0.f,
                         R[n][3], R[n][4], R[n][5], 0.f,
                         R[n][6], R[n][7], R[n][8], 0.f,
                         0.f, 0.f, 0.f, 1.f};
        float Ai[16];
        mat4mul(Ai, &L2P[n * 16], loc);
        float pose[16];
        if (n == 0) {
#pragma unroll
            for (int e = 0; e < 16; ++e) pose[e] = Ai[e];
        } else {
            mat4mul(pose, prev_pose, Ai);
        }
        float rest[16];
        if (n == 0) {
#pragma unroll
            for (int e = 0; e < 16; ++e) rest[e] = L2P[e];
        } else {
            mat4mul(rest, prev_rest, &L2P[n * 16]);
        }
        float W2L[16], WM[16];
        rigid_inv(W2L, rest);
        mat4mul(WM, pose, W2L);
#pragma unroll
        for (int e = 0; e < 12; ++e) Mws[b * 96 + n * 12 + e] = WM[e];
#pragma unroll
        for (int e = 0; e < 16; ++e) { prev_pose[e] = pose[e]; prev_rest[e] = rest[e]; }
    }
}

// GEMM: out[m, n] = T[n] + sum_k A[m,k] * Bmat[k,n]
// Bmat rows 0..71 = P (flattened [72, N3]); rows 72..126 = E; row 127 = 0.
// Wave32 per 16x16 tile; V_WMMA_F32_16X16X4_F32, 32 K-steps.
__global__ void __launch_bounds__(128)
k_gemm_wmma(const float* __restrict__ A,  // [256][128]
            const float* __restrict__ P,  // [72][N3]
            const float* __restrict__ E,  // [55][N3]
            const float* __restrict__ T,  // [N3]
            float* __restrict__ out)      // [256][N3]
{
    const int lane  = threadIdx.x & 31;
    const int wave  = threadIdx.x >> 5;
    const int ntile = blockIdx.x * 4 + wave;
    const int nbase = ntile * 16;
    if (nbase >= N3) return;  // wave-uniform exit: EXEC all-ones at WMMA

    const int mbase = blockIdx.y * 16;
    const int col   = lane & 15;
    const int khalf = lane >> 4;  // 0: K pair {0,1}; 1: K pair {2,3}

    const int n_idx   = nbase + col;
    const bool n_ok   = (n_idx < N3);
    const int n_clmp  = n_ok ? n_idx : (N3 - 1);

    const float* Arow = A + (mbase + col) * KP + khalf * 2;

    v8f acc = {};
#pragma unroll
    for (int kk = 0; kk < 32; ++kk) {
        const int k0 = kk * 4;
        v2f a;
        a.x = Arow[k0];
        a.y = Arow[k0 + 1];

        const int r0 = k0 + khalf * 2;   // 0..126
        const int r1 = r0 + 1;           // 1..127 (127 is the zero pad row)
        const int r1c = (r1 < 127) ? r1 : 126;
        const float* s0 = (r0 < 72) ? (P + (size_t)r0 * N3) : (E + (size_t)(r0 - 72) * N3);
        const float* s1 = (r1c < 72) ? (P + (size_t)r1c * N3) : (E + (size_t)(r1c - 72) * N3);
        float b0 = s0[n_clmp];
        float b1 = s1[n_clmp];
        b0 = n_ok ? b0 : 0.f;
        b1 = (n_ok && r1 < 127) ? b1 : 0.f;
        v2f bb; bb.x = b0; bb.y = b1;

        acc = __builtin_amdgcn_wmma_f32_16x16x4_f32(
            /*neg_a=*/false, a, /*neg_b=*/false, bb,
            /*c_mod=*/(short)0, acc, /*reuse_a=*/false, /*reuse_b=*/false);
    }

    const float tadd = n_ok ? T[n_idx] : 0.f;
#pragma unroll
    for (int r = 0; r < 8; ++r) {
        const int m = mbase + r + khalf * 8;  // C/D layout: lanes 16-31 -> M+8
        if (n_ok) out[(size_t)m * N3 + n_idx] = acc[r] + tadd;
    }
}

// Per-(b,v) epilogue: larynx bilinear sample + alpha*ts, then skinning matvec.
// In-place on d_out (same-stream ordering after k_gemm_wmma).
__global__ void __launch_bounds__(256)
k_final(const float* __restrict__ Wsk,   // [8][NV]
        const float* __restrict__ ts,    // [NV][3]
        const float* __restrict__ uvg,   // [NV][2]
        const float* __restrict__ Limg,  // [256][256]
        const float* __restrict__ tau,   // [256]
        const float* __restrict__ alpha, // [256]
        const float* __restrict__ Mws,   // [256][8][12]
        float* __restrict__ out)         // [256][NV][3]
{
    const int b = blockIdx.y;
    const int v = blockIdx.x * blockDim.x + threadIdx.x;

    __shared__ float sM[96];
    if (threadIdx.x < 96) sM[threadIdx.x] = Mws[b * 96 + threadIdx.x];
    __syncthreads();
    if (v >= NVTX) return;

    // bilinear sample, align_corners=True, border clamp (H=W=256)
    float gx = uvg[v * 2 + 0] * 2.f - 1.f;
    float gy = uvg[v * 2 + 1] * 2.f - 1.f + tau[b] * 2.f;
    float x = fminf(fmaxf((gx + 1.f) * 0.5f * 255.f, 0.f), 255.f);
    float y = fminf(fmaxf((gy + 1.f) * 0.5f * 255.f, 0.f), 255.f);
    float x0 = floorf(x), y0 = floorf(y);
    int x0i = (int)x0, y0i = (int)y0;
    int x1i = min(x0i + 1, 255), y1i = min(y0i + 1, 255);
    float wx = x - x0, wy = y - y0;
    float v00 = Limg[y0i * 256 + x0i], v01 = Limg[y0i * 256 + x1i];
    float v10 = Limg[y1i * 256 + x0i], v11 = Limg[y1i * 256 + x1i];
    float dist = (v00 * (1.f - wx) + v01 * wx) * (1.f - wy)
               + (v10 * (1.f - wx) + v11 * wx) * wy;
    float da = dist * alpha[b];

    const size_t base = ((size_t)b * NVTX + v) * 3;
    float t0 = out[base + 0] + da * ts[v * 3 + 0];
    float t1 = out[base + 1] + da * ts[v * 3 + 1];
    float t2 = out[base + 2] + da * ts[v * 3 + 2];

    float M[12];
#pragma unroll
    for (int e = 0; e < 12; ++e) M[e] = 0.f;
#pragma unroll
    for (int n = 0; n < NB; ++n) {
        float w = Wsk[n * NVTX + v];
#pragma unroll
        for (int e = 0; e < 12; ++e) M[e] = fmaf(sM[n * 12 + e], w, M[e]);
    }
    out[base + 0] = M[0] * t0 + M[1] * t1 + M[2]  * t2 + M[3];
    out[base + 1] = M[4] * t0 + M[5] * t1 + M[6]  * t2 + M[7];
    out[base + 2] = M[8] * t0 + M[9] * t1 + M[10] * t2 + M[11];
}

extern "C" void kernel_launch(void* const* d_in, const int* in_sizes, int n_in,
                              void* d_out, int out_size, void* d_ws, size_t ws_size,
                              hipStream_t stream) {
    const float* theta = (const float*)d_in[0];   // [256,8,3]
    const float* tau   = (const float*)d_in[1];   // [256,1]
    const float* alpha = (const float*)d_in[2];   // [256,1]
    const float* bsw   = (const float*)d_in[3];   // [256,55]
    const float* Wsk   = (const float*)d_in[4];   // [8,NV]
    const float* T     = (const float*)d_in[5];   // [NV,3]
    const float* P     = (const float*)d_in[6];   // [8,3,3,NV,3]
    const float* L     = (const float*)d_in[7];   // [1,1,256,256]
    const float* ts    = (const float*)d_in[8];   // [NV,3]
    const float* L2P   = (const float*)d_in[9];   // [8,4,4]
    const float* E     = (const float*)d_in[10];  // [55,NV,3]
    const float* uvg   = (const float*)d_in[11];  // [NV,2]
    float* out = (float*)d_out;

    float* A   = (float*)d_ws;            // 256*128 floats = 128 KB
    float* Mws = A + NBATCH * KP;         // 256*96 floats  = 96 KB

    k_precompute<<<1, NBATCH, 0, stream>>>(theta, bsw, L2P, A, Mws);

    const int ntiles = (N3 + 15) / 16;                // 2637
    dim3 g2((ntiles + 3) / 4, NBATCH / 16);           // 4 waves/block
    k_gemm_wmma<<<g2, 128, 0, stream>>>(A, P, E, T, out);

    dim3 g3((NVTX + 255) / 256, NBATCH);
    k_final<<<g3, 256, 0, stream>>>(Wsk, ts, uvg, L, tau, alpha, Mws, out);
}